// MyEmbedding_59390807769112
// MI455X (gfx1250) — compile-verified
//
#include <hip/hip_runtime.h>
#include <cstdint>
#include <cstddef>

#ifndef __has_builtin
#define __has_builtin(x) 0
#endif

#define HAS_ASYNC_LDS (__has_builtin(__builtin_amdgcn_global_load_async_to_lds_b128) && \
                       __has_builtin(__builtin_amdgcn_global_store_async_from_lds_b128) && \
                       __has_builtin(__builtin_amdgcn_s_wait_asynccnt))

// Pointer payload type the async-LDS b128 builtins expect (per hipcc diagnostic):
typedef int v4i __attribute__((vector_size(4 * sizeof(int))));
typedef __attribute__((address_space(3))) v4i lds_v4i;

// Problem geometry (fixed by the reference)
constexpr int kV     = 32000;   // base vocab rows
constexpr int kD     = 4096;    // embedding dim
constexpr int kB     = 2;       // batch
constexpr int kL     = 2048;    // sequence length
constexpr int kTImg  = 32;      // per-batch image latents
constexpr int kTReg  = 264;     // per-batch region latents (33*8)

constexpr int kThreads   = 256;                       // 8 waves (wave32)
constexpr int kRowBytes  = kD * 4;                    // 16 KiB per row
constexpr int kPassBytes = kThreads * 16;             // 4 KiB per b128 pass
constexpr int kPasses    = kRowBytes / kPassBytes;    // 4
constexpr int kTokensPerBlock = 4;                    // 64 KiB LDS, 16 async loads in flight
constexpr int kTotalTokens    = kB * kL;              // 4096
constexpr int kBlocks         = kTotalTokens / kTokensPerBlock;  // 1024

__device__ __forceinline__ const float* resolve_row(
    const float* __restrict__ weight,
    const float* __restrict__ img_tok,
    const float* __restrict__ reg_tok,
    const float* __restrict__ img_emb,
    const float* __restrict__ reg_emb,
    int idx, int b)
{
    if (idx < kV)                  return weight  + (size_t)idx * kD;
    if (idx < kV + 2)              return img_tok + (size_t)(idx - kV) * kD;
    if (idx < kV + 4)              return reg_tok + (size_t)(idx - kV - 2) * kD;
    if (idx < kV + 4 + kTImg)      return img_emb + ((size_t)b * kTImg + (size_t)(idx - kV - 4)) * kD;
    return reg_emb + ((size_t)b * kTReg + (size_t)(idx - kV - 4 - kTImg)) * kD;
}

__global__ __launch_bounds__(kThreads)
void embed_gather_async(const float* __restrict__ weight,
                        const float* __restrict__ img_tok,
                        const float* __restrict__ reg_tok,
                        const float* __restrict__ img_emb,
                        const float* __restrict__ reg_emb,
                        const int*   __restrict__ text,
                        float* __restrict__ out)
{
    __shared__ __align__(16) unsigned char smem[kTokensPerBlock][kRowBytes];

    const int token0    = blockIdx.x * kTokensPerBlock;
    const unsigned lane_off = threadIdx.x * 16u;

    const unsigned char* src8[kTokensPerBlock];
    #pragma unroll
    for (int t = 0; t < kTokensPerBlock; ++t) {
        const int token = token0 + t;
        const int b     = token / kL;
        const int idx   = text[token];   // block-uniform -> scalar load
        src8[t] = (const unsigned char*)resolve_row(weight, img_tok, reg_tok,
                                                    img_emb, reg_emb, idx, b);
    }

#if HAS_ASYNC_LDS
    // gfx1250 async data mover: burst 16 b128 async loads (4 tokens x 4 passes)
    // HBM -> LDS with no VGPR staging, then one ASYNCcnt drain, then the
    // mirrored store burst LDS -> HBM. Each wave touches only LDS its own
    // loads wrote, so no workgroup barrier is needed.
    #pragma unroll
    for (int t = 0; t < kTokensPerBlock; ++t) {
        #pragma unroll
        for (int k = 0; k < kPasses; ++k) {
            const unsigned off = lane_off + (unsigned)(k * kPassBytes);
            __builtin_amdgcn_global_load_async_to_lds_b128(
                (v4i*)(src8[t] + off),
                (lds_v4i*)(&smem[t][0] + off),
                /*offset=*/0, /*cpol=*/0);
        }
    }
    __builtin_amdgcn_s_wait_asynccnt(0);

    #pragma unroll
    for (int t = 0; t < kTokensPerBlock; ++t) {
        unsigned char* dst8 = (unsigned char*)(out + (size_t)(token0 + t) * kD);
        #pragma unroll
        for (int k = 0; k < kPasses; ++k) {
            const unsigned off = lane_off + (unsigned)(k * kPassBytes);
            __builtin_amdgcn_global_store_async_from_lds_b128(
                (v4i*)(dst8 + off),
                (lds_v4i*)(&smem[t][0] + off),
                /*offset=*/0, /*cpol=*/0);
        }
    }
    // S_ENDPGM does an implicit wait-idle; drain explicitly anyway so nothing
    // can be reordered past the store leg.
    __builtin_amdgcn_s_wait_asynccnt(0);
#else
    // Fallback: direct b128 copy through VGPRs.
    (void)smem;
    #pragma unroll
    for (int t = 0; t < kTokensPerBlock; ++t) {
        unsigned char* dst8 = (unsigned char*)(out + (size_t)(token0 + t) * kD);
        #pragma unroll
        for (int k = 0; k < kPasses; ++k) {
            const unsigned off = lane_off + (unsigned)(k * kPassBytes);
            *(float4*)(dst8 + off) = *(const float4*)(src8[t] + off);
        }
    }
#endif
}

extern "C" void kernel_launch(void* const* d_in, const int* in_sizes, int n_in,
                              void* d_out, int out_size, void* d_ws, size_t ws_size,
                              hipStream_t stream)
{
    const float* weight  = (const float*)d_in[0];  // [V, D]
    const float* img_tok = (const float*)d_in[1];  // [2, D]
    const float* reg_tok = (const float*)d_in[2];  // [2, D]
    const float* img_emb = (const float*)d_in[3];  // [B, T_IMG, D]
    const float* reg_emb = (const float*)d_in[4];  // [B, T_REG, D]
    const int*   text    = (const int*)d_in[5];    // [B, L] int32
    float*       out     = (float*)d_out;          // [B, L, D] f32

    (void)in_sizes; (void)n_in; (void)out_size; (void)d_ws; (void)ws_size;

    dim3 grid(kBlocks);      // 4 token rows per workgroup
    dim3 block(kThreads);
    hipLaunchKernelGGL(embed_gather_async, grid, block, 0, stream,
                       weight, img_tok, reg_tok, img_emb, reg_emb, text, out);
}